// SSSL_38697655337329
// MI455X (gfx1250) — compile-verified
//
#include <hip/hip_runtime.h>

// CDNA5 / gfx1250, wave32. V_WMMA_F32_16X16X4_F32 fp32 im2col-GEMM for the
// sliding-window cross-correlation; 4 t-tiles blocked per wave iteration so
// one B (shapelet) LDS load feeds 4 WMMAs. Online soft-min in registers.

typedef __attribute__((ext_vector_type(2))) float v2f;
typedef __attribute__((ext_vector_type(8))) float v8f;

#define SALPHA (-25.0f)
#define SH_STRIDE 205   // 205 mod 64 = 13, coprime with 64 -> conflict-free rows

__device__ __forceinline__ void softmerge(float& m, float& s, float& v,
                                          float m2, float s2, float v2) {
    float M  = fmaxf(m, m2);
    float e1 = __expf(m - M);
    float e2 = __expf(m2 - M);
    s = s * e1 + s2 * e2;
    v = v * e1 + v2 * e2;
    m = M;
}

__launch_bounds__(256)
__global__ void shapelet_softmin_kernel(const float* __restrict__ TS,
                                        const float* __restrict__ S,
                                        float* __restrict__ X) {
    __shared__ float ts[2112];            // TS row + zero pad (quad-tile overrun)
    __shared__ float c2[2049];            // prefix sums of squares
    __shared__ float T2[2048];            // sliding-window ||.||^2
    __shared__ float sh[32 * SH_STRIDE];  // shapelets, zero-padded K
    __shared__ float s2v[32];             // ||shapelet||^2
    __shared__ float part[256];           // scan scratch
    __shared__ float red[8 * 16 * 3];     // cross-wave softmax stats

    const int n   = blockIdx.x;       // series (0..511)
    const int r   = blockIdx.y;       // scale (0..3)
    const int l   = 50 * (r + 1);
    const int Tl  = 2048 - l + 1;
    const int tid = threadIdx.x;

    // ---- Stage 1: loads ----
    for (int i = tid; i < 2048; i += 256) ts[i] = TS[n * 2048 + i];
    for (int i = 2048 + tid; i < 2112; i += 256) ts[i] = 0.0f;

    const int lp = l + 4;
    for (int idx = tid; idx < 32 * lp; idx += 256) {
        int j = idx / lp, i = idx - j * lp;
        sh[j * SH_STRIDE + i] = (i < l) ? S[(r * 32 + j) * 200 + i] : 0.0f;
    }
    __syncthreads();

    // ---- Stage 2: prefix sums of squares -> T2; shapelet norms ----
    float loc[8];
    float sum = 0.0f;
#pragma unroll
    for (int i = 0; i < 8; ++i) {
        float x = ts[8 * tid + i];
        loc[i]  = x * x;
        sum    += loc[i];
    }
    part[tid] = sum;
    __syncthreads();
    for (int off = 1; off < 256; off <<= 1) {
        float add = (tid >= off) ? part[tid - off] : 0.0f;
        __syncthreads();
        part[tid] += add;
        __syncthreads();
    }
    float run = (tid == 0) ? 0.0f : part[tid - 1];
#pragma unroll
    for (int i = 0; i < 8; ++i) {
        run += loc[i];
        c2[8 * tid + i + 1] = run;
    }
    if (tid == 0) c2[0] = 0.0f;

    if (tid < 32) {
        float a = 0.0f;
        for (int i = 0; i < l; ++i) {
            float x = sh[tid * SH_STRIDE + i];
            a += x * x;
        }
        s2v[tid] = a;
    }
    __syncthreads();
    for (int t = tid; t < Tl; t += 256) T2[t] = c2[t + l] - c2[t];
    __syncthreads();

    // ---- Stage 3: WMMA cross-correlation, quad t-tile blocking ----
    const int wave = tid >> 5;
    const int lane = tid & 31;
    const int jt   = wave & 1;   // shapelet tile
    const int g    = wave >> 1;  // t-quad group (0..3)
    const int m16  = lane & 15;
    const int lg   = lane >> 4;  // K-pair select in A/B operand layout
    const int j    = jt * 16 + m16;
    const float s2j  = s2v[j];
    const float invl = 1.0f / (float)l;

    const int nT = (Tl + 15) >> 4;  // 16-row t-tiles
    const int nC = (l + 3) >> 2;    // K-chunks of 4 (zero-padded)

    float mA = -3.0e38f, sA = 0.0f, vA = 0.0f;

    for (int tb = 4 * g; tb < nT; tb += 16) {
        const int t0 = tb << 4;  // base t of quad (covers t0 .. t0+63)
        v8f acc[4] = {v8f{}, v8f{}, v8f{}, v8f{}};
        const float* aBase = &ts[t0 + m16 + 2 * lg];
        const float* bBase = &sh[j * SH_STRIDE + 2 * lg];
        for (int c = 0; c < nC; ++c) {
            v2f b;
            b.x = bBase[4 * c];
            b.y = bBase[4 * c + 1];
#pragma unroll
            for (int k = 0; k < 4; ++k) {
                v2f a;
                a.x = aBase[16 * k + 4 * c];
                a.y = aBase[16 * k + 4 * c + 1];
                acc[k] = __builtin_amdgcn_wmma_f32_16x16x4_f32(
                    false, a, false, b, (short)0, acc[k], false, false);
            }
        }
        // Epilogue: per tile, tile-max then single rescale (9 exps / 8 elems).
#pragma unroll
        for (int k = 0; k < 4; ++k) {
            float dv[8], av[8];
            float tmax = -3.0e38f;
#pragma unroll
            for (int v = 0; v < 8; ++v) {
                int t   = t0 + 16 * k + 8 * lg + v;
                float d = (t < Tl) ? (T2[t] - 2.0f * acc[k][v] + s2j) * invl
                                   : 1.0e30f;
                dv[v] = d;
                av[v] = SALPHA * d;
                tmax  = fmaxf(tmax, av[v]);
            }
            float M  = fmaxf(mA, tmax);
            float sc = __expf(mA - M);
            sA *= sc;
            vA *= sc;
#pragma unroll
            for (int v = 0; v < 8; ++v) {
                float w = __expf(av[v] - M);
                sA += w;
                vA += dv[v] * w;
            }
            mA = M;
        }
    }

    // merge lane pairs (same shapelet j): lane ^ 16
    {
        float m2 = __shfl_xor(mA, 16, 32);
        float s2 = __shfl_xor(sA, 16, 32);
        float v2 = __shfl_xor(vA, 16, 32);
        softmerge(mA, sA, vA, m2, s2, v2);
    }
    if (lg == 0) {
        red[(wave * 16 + m16) * 3 + 0] = mA;
        red[(wave * 16 + m16) * 3 + 1] = sA;
        red[(wave * 16 + m16) * 3 + 2] = vA;
    }
    __syncthreads();

    // ---- Stage 4: cross-wave merge, write X ----
    if (tid < 32) {
        int jj = tid & 15;
        float M = -3.0e38f, s = 0.0f, v = 0.0f;
        for (int w = (tid >> 4); w < 8; w += 2) {
            softmerge(M, s, v,
                      red[(w * 16 + jj) * 3 + 0],
                      red[(w * 16 + jj) * 3 + 1],
                      red[(w * 16 + jj) * 3 + 2]);
        }
        X[n * 128 + r * 32 + tid] = v / s;
    }
}

// Tiny dense layer: Z[n,c] = sum_k X[n,k] * W[c,k]  (512x10, 1.3 MFLOP)
__launch_bounds__(256)
__global__ void z_kernel(const float* __restrict__ X,
                         const float* __restrict__ W,
                         float* __restrict__ Z) {
    int i = blockIdx.x * 256 + threadIdx.x;
    if (i < 512 * 10) {
        int nn = i / 10, c = i % 10;
        float acc = 0.0f;
#pragma unroll 8
        for (int k = 0; k < 128; ++k) acc += X[nn * 128 + k] * W[c * 128 + k];
        Z[i] = acc;
    }
}

extern "C" void kernel_launch(void* const* d_in, const int* in_sizes, int n_in,
                              void* d_out, int out_size, void* d_ws, size_t ws_size,
                              hipStream_t stream) {
    const float* TS = (const float*)d_in[0];  // [512, 2048]
    const float* S  = (const float*)d_in[1];  // [128, 200]
    const float* W  = (const float*)d_in[2];  // [10, 128]
    float* X = (float*)d_out;                 // [512, 128]
    float* Z = X + 512 * 128;                 // [512, 10]

    dim3 grid(512, 4);
    shapelet_softmin_kernel<<<grid, 256, 0, stream>>>(TS, S, X);
    z_kernel<<<(512 * 10 + 255) / 256, 256, 0, stream>>>(X, W, Z);
}